// InformationLoss_59098749993405
// MI455X (gfx1250) — compile-verified
//
#include <hip/hip_runtime.h>
#include <math.h>

#define B_N 2048
#define R_N 64

typedef __attribute__((ext_vector_type(2))) float v2f;
typedef __attribute__((ext_vector_type(8))) float v8f;

constexpr float SQRT_2PI_F = 2.5066282746310002f;
constexpr float HCOEF = 0.23069590f;   // 1.06 * 2048^(-0.2)
constexpr float H_MIN_F = 1e-4f;
constexpr float EPS_LOG_F = 1e-8f;

// ---------------- K1: per-receptor mean / bandwidth ----------------
__global__ void il_stats_kernel(const float* __restrict__ act,
                                float* __restrict__ mean_out,
                                float* __restrict__ h_out) {
  __shared__ float s_sum[256];
  __shared__ float s_sq[256];
  const int r = blockIdx.x;
  const int tid = threadIdx.x;
  float s = 0.f, sq = 0.f;
  for (int q = tid; q < B_N; q += 256) {
    float v = act[q * R_N + r];
    s += v;
    sq += v * v;
  }
  s_sum[tid] = s; s_sq[tid] = sq;
  __syncthreads();
  for (int off = 128; off > 0; off >>= 1) {
    if (tid < off) { s_sum[tid] += s_sum[tid + off]; s_sq[tid] += s_sq[tid + off]; }
    __syncthreads();
  }
  if (tid == 0) {
    float sum = s_sum[0];
    float mu = sum / (float)B_N;
    float var = (s_sq[0] - sum * mu) / (float)(B_N - 1);   // ddof=1
    var = fmaxf(var, 0.f);
    mean_out[r] = mu;
    h_out[r] = fmaxf(HCOEF * sqrtf(var), H_MIN_F);
  }
}

// ---------------- K2: pairwise Gaussian KDE + log-density partials ----------------
__global__ void il_kde_kernel(const float* __restrict__ act,
                              const float* __restrict__ h_in,
                              float* __restrict__ partials) {
  __shared__ float col[B_N];   // one receptor column, 8 KB
  __shared__ float red[256];
  const int r = blockIdx.y;
  const int tid = threadIdx.x;

  for (int i = tid; i < B_N; i += 256) col[i] = act[i * R_N + r];
  __syncthreads();

  const int q = blockIdx.x * 256 + tid;
  const float x = col[q];
  const float inv_h = 1.0f / h_in[r];

  float acc = 0.f;
  const float4* c4 = (const float4*)col;
#pragma unroll 4
  for (int s4 = 0; s4 < B_N / 4; ++s4) {
    float4 v = c4[s4];
    float u0 = (x - v.x) * inv_h;
    float u1 = (x - v.y) * inv_h;
    float u2 = (x - v.z) * inv_h;
    float u3 = (x - v.w) * inv_h;
    acc += __expf(-0.5f * u0 * u0);
    acc += __expf(-0.5f * u1 * u1);
    acc += __expf(-0.5f * u2 * u2);
    acc += __expf(-0.5f * u3 * u3);
  }
  float density = acc * inv_h * (1.0f / (SQRT_2PI_F * (float)B_N));
  red[tid] = logf(density + EPS_LOG_F);
  __syncthreads();
  for (int off = 128; off > 0; off >>= 1) {
    if (tid < off) red[tid] += red[tid + off];
    __syncthreads();
  }
  if (tid == 0) partials[blockIdx.y * gridDim.x + blockIdx.x] = red[0];
}

// ---------------- K3: centered Gram (cov) via f32 WMMA, off-diag squared sum ----------------
__global__ void il_cov_wmma_kernel(const float* __restrict__ act,
                                   const float* __restrict__ mean_in,
                                   float* __restrict__ covsum) {
  __shared__ float xs[64 * 64];   // 64 K-rows x 64 receptors, centered (16 KB)
  __shared__ float red[512];
  const int tid  = threadIdx.x;
  const int lane = tid & 31;
  const int wave = tid >> 5;           // 16 waves, one 16x16 C tile each
  const int i0 = (wave >> 2) * 16;
  const int j0 = (wave & 3) * 16;
  const int half = lane >> 4;          // ISA 16x4 f32 A layout: half selects K {0,1} vs {2,3}
  const int m = lane & 15;

  v8f acc = {0.f, 0.f, 0.f, 0.f, 0.f, 0.f, 0.f, 0.f};

  for (int chunk = 0; chunk < B_N / 64; ++chunk) {
    __syncthreads();
    const int k0 = chunk * 64;
    for (int idx = tid; idx < 64 * 64; idx += 512) {   // exactly 8 iters, no divergence
      int kk = idx >> 6;
      int c = idx & 63;
      xs[idx] = act[(k0 + kk) * R_N + c] - mean_in[c];
    }
    __syncthreads();
#pragma unroll
    for (int kk = 0; kk < 64; kk += 4) {
      const int krow = kk + 2 * half;
      v2f A;  A.x  = xs[krow * 64 + i0 + m];  A.y  = xs[(krow + 1) * 64 + i0 + m];
      v2f Bv; Bv.x = xs[krow * 64 + j0 + m];  Bv.y = xs[(krow + 1) * 64 + j0 + m];
      acc = __builtin_amdgcn_wmma_f32_16x16x4_f32(
          /*neg_a=*/false, A, /*neg_b=*/false, Bv,
          /*c_mod=*/(short)0, acc, /*reuse_a=*/false, /*reuse_b=*/false);
    }
  }

  // C layout: VGPR v -> M = v + 8*half, N = lane&15
  const float inv = 1.0f / (float)(B_N - 1);
  float psum = 0.f;
#pragma unroll
  for (int v = 0; v < 8; ++v) {
    int i = i0 + v + half * 8;
    int j = j0 + m;
    float c = acc[v] * inv;
    if (i != j) psum += c * c;
  }
  red[tid] = psum;
  __syncthreads();
  for (int off = 256; off > 0; off >>= 1) {
    if (tid < off) red[tid] += red[tid + off];
    __syncthreads();
  }
  if (tid == 0) covsum[0] = red[0];
}

// ---------------- K4: deterministic finalize ----------------
__global__ void il_finalize_kernel(const float* __restrict__ partials,
                                   const float* __restrict__ covsum,
                                   float* __restrict__ out) {
  if (threadIdx.x == 0 && blockIdx.x == 0) {
    float s = 0.f;
    for (int i = 0; i < 512; ++i) s += partials[i];
    float loss_entropy = s * (1.0f / ((float)B_N * (float)R_N));
    out[0] = loss_entropy + covsum[0];
  }
}

extern "C" void kernel_launch(void* const* d_in, const int* in_sizes, int n_in,
                              void* d_out, int out_size, void* d_ws, size_t ws_size,
                              hipStream_t stream) {
  (void)in_sizes; (void)n_in; (void)out_size; (void)ws_size;
  const float* act = (const float*)d_in[0];
  float* ws = (float*)d_ws;
  float* mean_buf = ws;          // 64 floats
  float* h_buf    = ws + 64;     // 64 floats
  float* cov_buf  = ws + 128;    // 1 float
  float* partials = ws + 256;    // 512 floats (8 q-chunks x 64 receptors)

  il_stats_kernel<<<dim3(R_N), dim3(256), 0, stream>>>(act, mean_buf, h_buf);
  il_kde_kernel<<<dim3(8, R_N), dim3(256), 0, stream>>>(act, h_buf, partials);
  il_cov_wmma_kernel<<<dim3(1), dim3(512), 0, stream>>>(act, mean_buf, cov_buf);
  il_finalize_kernel<<<dim3(1), dim3(64), 0, stream>>>(partials, cov_buf, (float*)d_out);
}